// gGATLDA_91233695302377
// MI455X (gfx1250) — compile-verified
//
#include <hip/hip_runtime.h>
#include <math.h>

#define NNODES 100000
#define NEDGES 1600000
#define ETOT   (NEDGES + NNODES)
#define NEG_SLOPE 0.2f
#define TPB 256

typedef __attribute__((ext_vector_type(2))) float v2f;
typedef __attribute__((ext_vector_type(8))) float v8f;

__device__ __forceinline__ float eluf(float x)  { return x > 0.0f ? x : (expf(x) - 1.0f); }
__device__ __forceinline__ float lrelu(float x) { return x > 0.0f ? x : NEG_SLOPE * x; }

// Exact float atomic-max via sign-split integer atomics (IEEE order-preserving).
__device__ __forceinline__ void atomicMaxF(float* a, float v) {
    if (v >= 0.0f) atomicMax((int*)a, __float_as_int(v));
    else           atomicMin((unsigned int*)a, __float_as_uint(v));
}

__global__ void k_fill(float* __restrict__ p, int n, float v) {
    int i = blockIdx.x * blockDim.x + threadIdx.x;
    if (i < n) p[i] = v;
}

__global__ void k_build_edges(const int* __restrict__ ei, int* __restrict__ src,
                              int* __restrict__ dst) {
    int i = blockIdx.x * blockDim.x + threadIdx.x;
    if (i < NEDGES)    { src[i] = ei[i]; dst[i] = ei[NEDGES + i]; }
    else if (i < ETOT) { int v = i - NEDGES; src[i] = v; dst[i] = v; }
}

// Zero-pad B[K,N] into Bp[K,Npad] so the GEMM needs no column guards.
__global__ void k_pad_b(const float* __restrict__ B, float* __restrict__ Bp,
                        int K, int N, int Npad) {
    int i = blockIdx.x * blockDim.x + threadIdx.x;
    if (i >= K * Npad) return;
    int r = i / Npad, c = i - r * Npad;
    Bp[i] = (c < N) ? B[r * N + c] : 0.0f;
}

// C[M,Np] = A[M,K] @ B[K,Np]; Np is a multiple of 16; one wave per 16x16 C tile
// using V_WMMA_F32_16X16X4_F32. No guards anywhere -> branch-free inner loop.
// A 16x4 lane map: lanes 0-15 hold K=k,k+1; lanes 16-31 hold K=k+2,k+3 (ISA 7.12.2).
// B 4x16 mirrors it with N striped across lanes. C/D: VGPR r -> row m0+r(+8 hi half).
__global__ void k_gemm_wmma(const float* __restrict__ A, const float* __restrict__ B,
                            float* __restrict__ C, int M, int Np, int K) {
    int wave   = blockIdx.x * (blockDim.x >> 5) + (threadIdx.x >> 5);
    int tilesN = Np >> 4;
    int tiles  = (M >> 4) * tilesN;            // M is a multiple of 16 here
    if (wave >= tiles) return;
    int tm = wave / tilesN, tn = wave - tm * tilesN;
    int m0 = tm << 4, n0 = tn << 4;
    int lane = threadIdx.x & 31;
    int hi   = lane >> 4;
    int l16  = lane & 15;
    const float* ap = A + (size_t)(m0 + l16) * K;
    const float* bp = B + (size_t)n0 + l16;
    v8f acc = {};
    for (int k = 0; k < K; k += 4) {
        int kk = k + (hi << 1);
        v2f a, b;
        a.x = ap[kk];
        a.y = ap[kk + 1];
        b.x = bp[(size_t)kk * Np];
        b.y = bp[(size_t)(kk + 1) * Np];
        acc = __builtin_amdgcn_wmma_f32_16x16x4_f32(false, a, false, b,
                                                    (short)0, acc, false, false);
    }
    float* cp = C + (size_t)(m0 + (hi << 3)) * Np + n0 + l16;
#pragma unroll
    for (int r = 0; r < 8; ++r) cp[(size_t)r * Np] = acc[r];
}

// ---- GCN helpers ------------------------------------------------------------
__global__ void k_deg(const int* __restrict__ dst, float* __restrict__ deg) {
    int e = blockIdx.x * blockDim.x + threadIdx.x;
    if (e < ETOT) atomicAdd(&deg[dst[e]], 1.0f);
}

__global__ void k_dinv(float* __restrict__ deg) {
    int i = blockIdx.x * blockDim.x + threadIdx.x;
    if (i < NNODES) { float d = deg[i]; deg[i] = d > 0.0f ? rsqrtf(d) : 0.0f; }
}

// out[dst] += h[src] * dinv[src]*dinv[dst], 16 feats, one thread per (edge, 4-feat chunk)
__global__ void k_gcn_agg(const int* __restrict__ src, const int* __restrict__ dst,
                          const float* __restrict__ dinv, const float* __restrict__ h,
                          float* __restrict__ out) {
    long i = (long)blockIdx.x * blockDim.x + threadIdx.x;
    if (i >= (long)ETOT * 4) return;
    int e = (int)(i >> 2), c = (int)(i & 3);
    int s = src[e], d = dst[e];
    float norm = dinv[s] * dinv[d];
    float4 v = *(const float4*)(h + (size_t)s * 16 + c * 4);
    float* o = out + (size_t)d * 16 + c * 4;
    atomicAdd(o + 0, v.x * norm); atomicAdd(o + 1, v.y * norm);
    atomicAdd(o + 2, v.z * norm); atomicAdd(o + 3, v.w * norm);
}

// ---- GAT helpers ------------------------------------------------------------
// s[n,h] = <hW[n,h,:], a_src[h,:]> ; t likewise. hW row stride = ldH (padded).
__global__ void k_scores(const float* __restrict__ hW, const float* __restrict__ as,
                         const float* __restrict__ ad, float* __restrict__ s,
                         float* __restrict__ t, int H, int C, int ldH) {
    int i = blockIdx.x * blockDim.x + threadIdx.x;
    if (i >= NNODES * H) return;
    int n = i / H, h = i - n * H;
    const float* hp = hW + (size_t)n * ldH + h * C;
    const float* ap = as + (size_t)h * C;
    const float* dp = ad + (size_t)h * C;
    float ss = 0.0f, tt = 0.0f;
    for (int j = 0; j < C; ++j) { float v = hp[j]; ss += v * ap[j]; tt += v * dp[j]; }
    s[i] = ss; t[i] = tt;
}

__global__ void k_edge_max(const int* __restrict__ src, const int* __restrict__ dst,
                           const float* __restrict__ s, const float* __restrict__ t,
                           float* __restrict__ m, int H) {
    long i = (long)blockIdx.x * blockDim.x + threadIdx.x;
    if (i >= (long)ETOT * H) return;
    int e = (int)(i / H), h = (int)(i - (long)e * H);
    int sn = src[e], dn = dst[e];
    float x = lrelu(s[sn * H + h] + t[dn * H + h]);
    atomicMaxF(&m[dn * H + h], x);
}

__global__ void k_edge_sum(const int* __restrict__ src, const int* __restrict__ dst,
                           const float* __restrict__ s, const float* __restrict__ t,
                           const float* __restrict__ m, float* __restrict__ denom, int H) {
    long i = (long)blockIdx.x * blockDim.x + threadIdx.x;
    if (i >= (long)ETOT * H) return;
    int e = (int)(i / H), h = (int)(i - (long)e * H);
    int sn = src[e], dn = dst[e];
    float x = lrelu(s[sn * H + h] + t[dn * H + h]);
    atomicAdd(&denom[dn * H + h], expf(x - m[dn * H + h]));
}

// out[dst,h,:] += hW[src,h,:] * alpha; one thread per (edge, head). hW stride ldH.
__global__ void k_edge_agg(const int* __restrict__ src, const int* __restrict__ dst,
                           const float* __restrict__ s, const float* __restrict__ t,
                           const float* __restrict__ m, const float* __restrict__ denom,
                           const float* __restrict__ hW, float* __restrict__ out,
                           int H, int C, int ldH) {
    long i = (long)blockIdx.x * blockDim.x + threadIdx.x;
    if (i >= (long)ETOT * H) return;
    int e = (int)(i / H), h = (int)(i - (long)e * H);
    int sn = src[e], dn = dst[e];
    int idx = dn * H + h;
    float x = lrelu(s[sn * H + h] + t[idx]);
    float alpha = expf(x - m[idx]) / (denom[idx] + 1e-16f);
    int HC = H * C;
    const float* hp = hW + (size_t)sn * ldH + h * C;
    float* op = out + (size_t)dn * HC + h * C;
    if ((C & 3) == 0) {
        for (int j = 0; j < C; j += 4) {
            float4 v = *(const float4*)(hp + j);
            atomicAdd(op + j + 0, v.x * alpha); atomicAdd(op + j + 1, v.y * alpha);
            atomicAdd(op + j + 2, v.z * alpha); atomicAdd(op + j + 3, v.w * alpha);
        }
    } else {
        for (int j = 0; j < C; ++j) atomicAdd(op + j, hp[j] * alpha);
    }
}

__global__ void k_bias_elu(float* __restrict__ p, const float* __restrict__ b,
                           int F, int total) {
    int i = blockIdx.x * blockDim.x + threadIdx.x;
    if (i >= total) return;
    int f = i % F;
    p[i] = eluf(p[i] + b[f]);
}

// out[i,:] = log_softmax(concat(h[i], h[half+i])) for i in [0, half)
__global__ void k_out(const float* __restrict__ h, float* __restrict__ out) {
    int i = blockIdx.x * blockDim.x + threadIdx.x;
    const int half = NNODES / 2;
    if (i >= half) return;
    float v0 = h[(size_t)i * 2], v1 = h[(size_t)i * 2 + 1];
    float v2 = h[(size_t)(half + i) * 2], v3 = h[(size_t)(half + i) * 2 + 1];
    float mx = fmaxf(fmaxf(v0, v1), fmaxf(v2, v3));
    float lse = mx + logf(expf(v0 - mx) + expf(v1 - mx) + expf(v2 - mx) + expf(v3 - mx));
    out[(size_t)i * 4 + 0] = v0 - lse; out[(size_t)i * 4 + 1] = v1 - lse;
    out[(size_t)i * 4 + 2] = v2 - lse; out[(size_t)i * 4 + 3] = v3 - lse;
}

extern "C" void kernel_launch(void* const* d_in, const int* in_sizes, int n_in,
                              void* d_out, int out_size, void* d_ws, size_t ws_size,
                              hipStream_t stream) {
    (void)in_sizes; (void)n_in; (void)out_size; (void)ws_size;
    const float* x    = (const float*)d_in[0];
    const int*   ei   = (const int*)d_in[1];
    const float* gcnW = (const float*)d_in[2];
    const float* gcnB = (const float*)d_in[3];
    const float* gW[4]  = {(const float*)d_in[4], (const float*)d_in[8],
                           (const float*)d_in[12], (const float*)d_in[16]};
    const float* gAS[4] = {(const float*)d_in[5], (const float*)d_in[9],
                           (const float*)d_in[13], (const float*)d_in[17]};
    const float* gAD[4] = {(const float*)d_in[6], (const float*)d_in[10],
                           (const float*)d_in[14], (const float*)d_in[18]};
    const float* gB[4]  = {(const float*)d_in[7], (const float*)d_in[11],
                           (const float*)d_in[15], (const float*)d_in[19]};

    char* ws = (char*)d_ws;
    size_t off = 0;
    auto carve = [&](size_t bytes) {
        char* p = ws + off;
        off += (bytes + 255) & ~(size_t)255;
        return p;
    };
    int*   src  = (int*)  carve(sizeof(int)   * (size_t)ETOT);
    int*   dst  = (int*)  carve(sizeof(int)   * (size_t)ETOT);
    float* bufA = (float*)carve(sizeof(float) * (size_t)NNODES * 128);  // h = in @ W
    float* bufB = (float*)carve(sizeof(float) * (size_t)NNODES * 128);  // layer output
    float* sb   = (float*)carve(sizeof(float) * (size_t)NNODES * 8);
    float* tb   = (float*)carve(sizeof(float) * (size_t)NNODES * 8);
    float* mb   = (float*)carve(sizeof(float) * (size_t)NNODES * 8);
    float* db   = (float*)carve(sizeof(float) * (size_t)NNODES * 8);
    float* deg  = (float*)carve(sizeof(float) * (size_t)NNODES);
    float* Bpad = (float*)carve(sizeof(float) * 128 * 128);             // padded weights

    auto nb = [](long n) { return dim3((unsigned)((n + TPB - 1) / TPB)); };
    // Returns the (padded) leading dimension of C.
    auto gemm = [&](const float* A, const float* B, float* C, int M, int N, int K) {
        int Np = (N + 15) & ~15;
        const float* Bu = B;
        if (Np != N) {
            k_pad_b<<<nb((long)K * Np), TPB, 0, stream>>>(B, Bpad, K, N, Np);
            Bu = Bpad;
        }
        int tiles = (M / 16) * (Np / 16);
        dim3 grid((unsigned)((tiles + 3) / 4));   // 4 waves / 128-thread block
        k_gemm_wmma<<<grid, dim3(128), 0, stream>>>(A, Bu, C, M, Np, K);
        return Np;
    };

    k_build_edges<<<nb(ETOT), TPB, 0, stream>>>(ei, src, dst);

    // ---- GCN layer (out: bufB, 16 feats) ----
    gemm(x, gcnW, bufA, NNODES, 16, 128);
    k_fill<<<nb(NNODES), TPB, 0, stream>>>(deg, NNODES, 0.0f);
    k_deg<<<nb(ETOT), TPB, 0, stream>>>(dst, deg);
    k_dinv<<<nb(NNODES), TPB, 0, stream>>>(deg);
    k_fill<<<nb((long)NNODES * 16), TPB, 0, stream>>>(bufB, NNODES * 16, 0.0f);
    k_gcn_agg<<<nb((long)ETOT * 4), TPB, 0, stream>>>(src, dst, deg, bufA, bufB);
    k_bias_elu<<<nb((long)NNODES * 16), TPB, 0, stream>>>(bufB, gcnB, 16, NNODES * 16);

    // ---- GAT layers (in: bufB, out: bufB) ----
    auto gat = [&](int L, int F, int H, int C) {
        int HC = H * C;
        int ldH = gemm(bufB, gW[L], bufA, NNODES, HC, F);   // bufA = h, stride ldH
        k_scores<<<nb((long)NNODES * H), TPB, 0, stream>>>(bufA, gAS[L], gAD[L],
                                                           sb, tb, H, C, ldH);
        k_fill<<<nb((long)NNODES * H), TPB, 0, stream>>>(mb, NNODES * H, -1.0e30f);
        k_fill<<<nb((long)NNODES * H), TPB, 0, stream>>>(db, NNODES * H, 0.0f);
        k_fill<<<nb((long)NNODES * HC), TPB, 0, stream>>>(bufB, NNODES * HC, 0.0f);
        k_edge_max<<<nb((long)ETOT * H), TPB, 0, stream>>>(src, dst, sb, tb, mb, H);
        k_edge_sum<<<nb((long)ETOT * H), TPB, 0, stream>>>(src, dst, sb, tb, mb, db, H);
        k_edge_agg<<<nb((long)ETOT * H), TPB, 0, stream>>>(src, dst, sb, tb, mb, db,
                                                           bufA, bufB, H, C, ldH);
        k_bias_elu<<<nb((long)NNODES * HC), TPB, 0, stream>>>(bufB, gB[L], HC, NNODES * HC);
    };
    gat(0, 16, 8, 16);
    gat(1, 128, 8, 16);
    gat(2, 128, 8, 16);
    gat(3, 128, 1, 2);

    k_out<<<nb(NNODES / 2), TPB, 0, stream>>>(bufB, (float*)d_out);
}